// Backflow_75179107549498
// MI455X (gfx1250) — compile-verified
//
#include <hip/hip_runtime.h>
#include <hip/hip_bf16.h>
#include <math.h>

typedef __attribute__((ext_vector_type(2))) float v2f;
typedef __attribute__((ext_vector_type(8))) float v8f;

#define NB 1024
#define NORB 128
#define MM 64
#define TT 8
#define LDSTRIDE 65   // 64 + 1 pad: conflict-free rows AND columns (65 mod 64 = 1)

#define A_(r,c_) As[(r)*LDSTRIDE + (c_)]

// One wave32 per (b, spin, t): gather 64x64 f32 matrix, blocked LU with
// partial pivoting (16-wide panels), WMMA f32 16x16x4 trailing updates.
// Writes (logdet, sign) per matrix into ws.
__global__ __launch_bounds__(32) void lu_slogdet_kernel(
    const float* __restrict__ orbitals,      // [NORB][MM]
    const float* __restrict__ corr,          // [B][NORB][MM][TT]
    const int*   __restrict__ y_up,          // [B][MM]
    const int*   __restrict__ y_dn,          // [B][MM]
    float* __restrict__ ws)                  // [2][B][TT][2] (logdet, sign)
{
    __shared__ float As[MM * LDSTRIDE];

    const int lane = threadIdx.x;            // 0..31 (wave32)
    const int b    = blockIdx.y;
    const int st   = blockIdx.x;             // 0..15
    const int spin = st >> 3;
    const int t    = st & 7;
    const int* __restrict__ y = spin ? y_dn : y_up;

    // ---- gather: A[m][n] = orbitals[o][n] + corr[b][o][n][t], o=(y[b][m]+16t)%128
    // Row-cooperative: all 32 lanes on one row -> each vmem instruction spans a
    // contiguous 1KB region (32B stride across lanes) for clean coalescing;
    // y[] and row bases become uniform scalar loads.
    for (int m = 0; m < MM; ++m) {
        // prefetch the corrections row we will need 4 iterations from now
        if (m + 4 < MM) {
            const int o2 = (y[b * MM + m + 4] + 16 * t) & (NORB - 1);
            const float* p = corr + ((size_t)(b * NORB + o2) * MM) * TT + t;
            __builtin_prefetch(p + (size_t)lane * 2 * TT, 0, 1);  // covers 2KB row block
        }
        const int o = (y[b * MM + m] + 16 * t) & (NORB - 1);
        const float* __restrict__ orow = orbitals + o * MM;
        const float* __restrict__ crow = corr + ((size_t)(b * NORB + o) * MM) * TT + t;
        A_(m, lane)      = orow[lane]      + crow[(size_t)lane * TT];
        A_(m, lane + 32) = orow[lane + 32] + crow[(size_t)(lane + 32) * TT];
    }
    __syncthreads();

    float logdet = 0.0f;
    float sign   = 1.0f;

    const int halfsel = lane >> 4;   // 0: lanes 0-15, 1: lanes 16-31
    const int lid     = lane & 15;

    for (int k = 0; k < 4; ++k) {
        const int c0 = k * 16;

        // ---------- panel factorization: cols c0..c0+15, rows c..63 ----------
        for (int j = 0; j < 16; ++j) {
            const int c = c0 + j;
            __syncthreads();
            // partial-pivot search down column c
            float bv = -1.0f; int br = c;
            for (int r = c + lane; r < MM; r += 32) {
                const float v = fabsf(A_(r, c));
                if (v > bv) { bv = v; br = r; }
            }
            #pragma unroll
            for (int off = 16; off >= 1; off >>= 1) {
                const float ov = __shfl_xor(bv, off, 32);
                const int   orr = __shfl_xor(br, off, 32);
                if (ov > bv || (ov == bv && orr < br)) { bv = ov; br = orr; }
            }
            __syncthreads();
            if (br != c) {             // uniform branch; swap full rows
                for (int col = lane; col < MM; col += 32) {
                    const float tmp = A_(c, col);
                    A_(c, col)  = A_(br, col);
                    A_(br, col) = tmp;
                }
                sign = -sign;
            }
            __syncthreads();
            const float piv = A_(c, c);
            logdet += logf(fabsf(piv));
            if (piv < 0.0f) sign = -sign;
            const float rinv = 1.0f / piv;
            // broadcast pivot-row panel segment into registers
            float urow[16];
            for (int jj = j + 1; jj < 16; ++jj) urow[jj] = A_(c, c0 + jj);
            __syncthreads();
            // scale multipliers + rank-1 update restricted to panel
            for (int r = c + 1 + lane; r < MM; r += 32) {
                const float l = A_(r, c) * rinv;
                A_(r, c) = l;
                for (int jj = j + 1; jj < 16; ++jj)
                    A_(r, c0 + jj) -= l * urow[jj];
            }
        }
        __syncthreads();
        if (k == 3) break;

        // ---------- U12 solve: L11 (unit-lower) * U12 = A12 ----------
        for (int j = 1; j < 16; ++j) {
            const int rj = c0 + j;
            for (int col = c0 + 16 + lane; col < MM; col += 32) {
                float acc = A_(rj, col);
                for (int p = 0; p < j; ++p)
                    acc -= A_(rj, c0 + p) * A_(c0 + p, col);
                A_(rj, col) = acc;
            }
        }
        __syncthreads();

        // ---------- trailing update A22 -= L21 * U12 via WMMA f32 ----------
        // (no lane divergence: EXEC must be all-1s for WMMA)
        const int nt = 3 - k;
        for (int ti = 0; ti < nt; ++ti) {
            const int I = c0 + 16 + ti * 16;
            for (int tj = 0; tj < nt; ++tj) {
                const int J = c0 + 16 + tj * 16;
                v8f acc = {};
                #pragma unroll
                for (int kk = 0; kk < 4; ++kk) {
                    // A frag 16x4 (ISA 7.12.2): lanes 0-15 -> K=kk*4+{0,1},
                    // lanes 16-31 -> K=kk*4+{2,3}; M = lid
                    v2f af, bf;
                    const int akc = c0 + kk * 4 + halfsel * 2;
                    af.x = A_(I + lid, akc);
                    af.y = A_(I + lid, akc + 1);
                    // B frag 4x16: rows striped across lanes within a VGPR
                    const int bkr = c0 + kk * 4 + halfsel * 2;
                    bf.x = A_(bkr,     J + lid);
                    bf.y = A_(bkr + 1, J + lid);
                    acc = __builtin_amdgcn_wmma_f32_16x16x4_f32(
                        false, af, false, bf, (short)0, acc, false, false);
                }
                // C/D layout: VGPR r -> M = r + 8*halfsel, N = lid
                #pragma unroll
                for (int r = 0; r < 8; ++r)
                    A_(I + r + halfsel * 8, J + lid) -= acc[r];
            }
        }
        __syncthreads();
    }

    if (lane == 0) {
        const int idx = ((spin * NB + b) * TT + t) * 2;
        ws[idx]     = logdet;
        ws[idx + 1] = sign;
    }
}

// Per-batch complex log-sum-exp over T combining up/dn spins.
__global__ void combine_kernel(const float* __restrict__ ws,
                               float* __restrict__ out)   // complex64 -> interleaved
{
    const int b = blockIdx.x * blockDim.x + threadIdx.x;
    if (b >= NB) return;

    float re[TT], sg[TT];
    float mx = -INFINITY;
    #pragma unroll
    for (int t = 0; t < TT; ++t) {
        const int iu = ((0 * NB + b) * TT + t) * 2;
        const int id = ((1 * NB + b) * TT + t) * 2;
        re[t] = ws[iu] + ws[id];
        sg[t] = ws[iu + 1] * ws[id + 1];
        mx = fmaxf(mx, re[t]);
    }
    const float PI_F = 3.14159265358979323846f;
    float sr = 0.0f, si = 0.0f;
    #pragma unroll
    for (int t = 0; t < TT; ++t) {
        const float e  = expf(re[t] - mx);
        const float im = (sg[t] < 0.0f) ? PI_F : 0.0f;  // log(sign + 0j).imag
        sr += e * cosf(im);
        si += e * sinf(im);
    }
    out[2 * b]     = logf(sqrtf(sr * sr + si * si)) + mx;
    out[2 * b + 1] = atan2f(si, sr);
}

extern "C" void kernel_launch(void* const* d_in, const int* in_sizes, int n_in,
                              void* d_out, int out_size, void* d_ws, size_t ws_size,
                              hipStream_t stream) {
    const float* orbitals = (const float*)d_in[0];   // [128][64] f32
    const float* corr     = (const float*)d_in[1];   // [1024][128][64][8] f32
    const int*   y_up     = (const int*)d_in[2];     // [1024][64] i32
    const int*   y_dn     = (const int*)d_in[3];     // [1024][64] i32
    float* ws  = (float*)d_ws;                       // 2*1024*8*2 floats = 128 KB
    float* out = (float*)d_out;                      // 1024 complex64

    // b is the slow grid index: the 16 matrices sharing corrections[b]
    // are adjacent blocks -> L2 reuse of the 256 KB corrections slice.
    dim3 grid(16, NB);
    lu_slogdet_kernel<<<grid, 32, 0, stream>>>(orbitals, corr, y_up, y_dn, ws);
    combine_kernel<<<(NB + 255) / 256, 256, 0, stream>>>(ws, out);
}